// DA_affinity_attention_72653666779471
// MI455X (gfx1250) — compile-verified
//
#include <hip/hip_runtime.h>
#include <hip/hip_bf16.h>
#include <math.h>

// Problem constants (from reference)
#define B_    2
#define NQ_   512
#define NKV_  2048
#define C_    32      // ERP_DIM
#define ICO_  64
#define ITERS_ 2
#define L2E_  1.44269504088896340736f

typedef float v2f __attribute__((ext_vector_type(2)));
typedef float v8f __attribute__((ext_vector_type(8)));
typedef int   v4i __attribute__((vector_size(4 * sizeof(int))));

// ---------------------------------------------------------------------------
// gfx1250 async global->LDS copy (ASYNCcnt path), with safe fallback.
// Builtin signature (from hipcc diagnostic): (AS1 v4i* src, AS3 v4i* dst,
// imm int offset, imm int cpol).
// ---------------------------------------------------------------------------
__device__ __forceinline__ void async_copy16(const void* g, void* l) {
#if __has_builtin(__builtin_amdgcn_global_load_async_to_lds_b128)
    __builtin_amdgcn_global_load_async_to_lds_b128(
        (__attribute__((address_space(1))) v4i*)g,
        (__attribute__((address_space(3))) v4i*)l, 0, 0);
#else
    *(float4*)l = *(const float4*)g;
#endif
}
__device__ __forceinline__ void async_wait0() {
#if __has_builtin(__builtin_amdgcn_s_wait_asynccnt)
    __builtin_amdgcn_s_wait_asynccnt(0);
#endif
}

// ---------------------------------------------------------------------------
// Generic fp32 GEMM tile via V_WMMA_F32_16X16X4_F32 (one 16x16 tile / wave).
// Columns < scale_cols get multiplied by log2(e) in the epilogue so the
// affinity kernel's inner loop needs no multiply (exp2(-|d|) folds the
// neg/abs into VALU source modifiers).
// ---------------------------------------------------------------------------
__global__ void proj_gemm_kernel(const float* __restrict__ In,
                                 const float* __restrict__ W,
                                 const float* __restrict__ bias,
                                 float* __restrict__ Out,
                                 int R, int K, int COUT, int scale_cols)
{
    const int lane = threadIdx.x & 31;
    const int warp = threadIdx.x >> 5;
    const int tiles_c = COUT >> 4;
    const int n_tiles = (R >> 4) * tiles_c;
    int tile = blockIdx.x * (blockDim.x >> 5) + warp;
    if (tile >= n_tiles) return;
    const int rt = tile / tiles_c;
    const int ct = tile - rt * tiles_c;
    const int r0 = rt << 4, c0 = ct << 4;
    const int half = lane >> 4;
    const int rr = lane & 15;

    v8f acc = {};
    for (int kk = 0; kk < K; kk += 4) {
        v2f a, b;
        const float* arow = In + (size_t)(r0 + rr) * K + kk + 2 * half;
        a.x = arow[0];
        a.y = arow[1];
        const float* brow = W + (size_t)(c0 + rr) * K + kk + 2 * half;
        b.x = brow[0];
        b.y = brow[1];
        acc = __builtin_amdgcn_wmma_f32_16x16x4_f32(
            false, a, false, b, (short)0, acc, false, false);
    }
    const float alpha = ((c0 + rr) < scale_cols) ? L2E_ : 1.0f;
    const float bv = bias ? bias[c0 + rr] : 0.0f;
#pragma unroll
    for (int g = 0; g < 8; ++g)
        Out[(size_t)(r0 + g + 8 * half) * COUT + c0 + rr] = acc[g] * alpha + bv;
}

// ---------------------------------------------------------------------------
// Affinity: S[b,n,m] = sum_c 2^(-|q^[n,c]-k^[m,c]|)
//                    + sum_t wsum[t] * 2^(-|L2E*(qc[n,t]-kc[m,t])|)
// q^ / k^ are pre-scaled by log2(e); coords scaled at staging time.
// wsum[t] = sum_c Wdelta[c,t] (pos_enc reduced analytically).
// Block = 256 threads; tile = 16 queries x 256 keys; keys async-staged in LDS.
// ---------------------------------------------------------------------------
__global__ void affinity_kernel(const float* __restrict__ qp,    // (B*NQ, C), * L2E
                                const float* __restrict__ qc,    // (B*NQ, 3)
                                const float* __restrict__ kvp,   // (B*NKV, 2C), k=[0,C) * L2E
                                const float* __restrict__ kc,    // (B*NKV, 3)
                                const float* __restrict__ Wd,    // (C, 3)
                                float* __restrict__ S)           // (B*NQ, NKV)
{
    __shared__ float4 sK[256 * 8];   // 256 keys x 32 ch
    __shared__ float4 sQ[16 * 8];    // 16 queries x 32 ch
    __shared__ float  sKC[256 * 3];
    __shared__ float  sQC[16 * 3];
    __shared__ float  sW[3];

    const int t = threadIdx.x;
    const int MT = NKV_ / 256;
    int bid = blockIdx.x;
    const int mt = bid % MT;         bid /= MT;
    const int nt = bid % (NQ_ / 16); bid /= (NQ_ / 16);
    const int b  = bid;
    const int n0 = nt * 16, m0 = mt * 256;

    // async-stage key features (2048 x 16B) and query features (128 x 16B)
#pragma unroll
    for (int j = 0; j < 8; ++j) {
        const int idx = t + j * 256;
        const int m = idx >> 3, c4 = idx & 7;
        async_copy16(kvp + (size_t)(b * NKV_ + m0 + m) * (2 * C_) + c4 * 4, &sK[idx]);
    }
    if (t < 128) {
        const int i = t >> 3, c4 = t & 7;
        async_copy16(qp + (size_t)(b * NQ_ + n0 + i) * C_ + c4 * 4, &sQ[t]);
    }
    // scalar-staged (scaled) coordinates + wsum overlap the async fills
    if (t < 3) {
        float s = 0.0f;
        for (int c = 0; c < C_; ++c) s += Wd[c * 3 + t];
        sW[t] = s;
    }
    if (t < 48) {
        const int i = t / 3, t3 = t % 3;
        sQC[t] = qc[(size_t)(b * NQ_ + n0 + i) * 3 + t3] * L2E_;
    }
#pragma unroll
    for (int j = 0; j < 3; ++j) {
        const int idx = t + j * 256;
        const int m = idx / 3, t3 = idx % 3;
        sKC[idx] = kc[(size_t)(b * NKV_ + m0 + m) * 3 + t3] * L2E_;
    }
    async_wait0();
    __syncthreads();

    const int nl  = t >> 4;   // query 0..15
    const int ml0 = t & 15;

    float4 qr[8];
#pragma unroll
    for (int c4 = 0; c4 < 8; ++c4) qr[c4] = sQ[nl * 8 + c4];
    const float q0 = sQC[nl * 3 + 0], q1 = sQC[nl * 3 + 1], q2 = sQC[nl * 3 + 2];
    const float w0 = sW[0], w1 = sW[1], w2 = sW[2];

    float* srow = S + (size_t)(b * NQ_ + n0 + nl) * NKV_ + m0;
#pragma unroll 4
    for (int j = 0; j < 16; ++j) {
        const int ml = ml0 + j * 16;
        float acc = 0.0f;
#pragma unroll
        for (int c4 = 0; c4 < 8; ++c4) {
            const float4 k4 = sK[ml * 8 + c4];
            const float4 q4 = qr[c4];
            acc += __builtin_amdgcn_exp2f(-fabsf(q4.x - k4.x));  // sub + exp w/ mods
            acc += __builtin_amdgcn_exp2f(-fabsf(q4.y - k4.y));
            acc += __builtin_amdgcn_exp2f(-fabsf(q4.z - k4.z));
            acc += __builtin_amdgcn_exp2f(-fabsf(q4.w - k4.w));
        }
        acc += w0 * __builtin_amdgcn_exp2f(-fabsf(q0 - sKC[ml * 3 + 0]));
        acc += w1 * __builtin_amdgcn_exp2f(-fabsf(q1 - sKC[ml * 3 + 1]));
        acc += w2 * __builtin_amdgcn_exp2f(-fabsf(q2 - sKC[ml * 3 + 2]));
        srow[ml] = acc;
    }
}

// ---------------------------------------------------------------------------
// In-place row softmax of S*(1/C); one wave32 per row; each lane keeps its
// contiguous 64 values in registers -> single read + single write of S.
// ---------------------------------------------------------------------------
__global__ void softmax_kernel(float* __restrict__ S)
{
    const int lane = threadIdx.x & 31;
    const int warp = threadIdx.x >> 5;
    const int row  = blockIdx.x * (blockDim.x >> 5) + warp;
    if (row >= B_ * NQ_) return;
    float4* p = (float4*)(S + (size_t)row * NKV_) + lane * (NKV_ / 32 / 4);

    float4 v[16];
    float mx = -3.402823e38f;
#pragma unroll
    for (int i = 0; i < 16; ++i) {
        v[i] = p[i];
        mx = fmaxf(mx, fmaxf(fmaxf(v[i].x, v[i].y), fmaxf(v[i].z, v[i].w)));
    }
    for (int off = 16; off; off >>= 1) mx = fmaxf(mx, __shfl_xor(mx, off, 32));

    const float sc = (1.0f / C_) * L2E_;
    float sum = 0.0f;
#pragma unroll
    for (int i = 0; i < 16; ++i) {
        v[i].x = __builtin_amdgcn_exp2f((v[i].x - mx) * sc);
        v[i].y = __builtin_amdgcn_exp2f((v[i].y - mx) * sc);
        v[i].z = __builtin_amdgcn_exp2f((v[i].z - mx) * sc);
        v[i].w = __builtin_amdgcn_exp2f((v[i].w - mx) * sc);
        sum += v[i].x + v[i].y + v[i].z + v[i].w;
    }
    for (int off = 16; off; off >>= 1) sum += __shfl_xor(sum, off, 32);
    const float inv = 1.0f / sum;

#pragma unroll
    for (int i = 0; i < 16; ++i) {
        v[i].x *= inv; v[i].y *= inv; v[i].z *= inv; v[i].w *= inv;
        p[i] = v[i];
    }
}

// ---------------------------------------------------------------------------
// qout[n,c] = sum_m attn[n,m] * v[m,c]. Block owns one (batch, c-tile) and
// 8 n-tiles (one per wave); v is async-staged chunk-wise into LDS and shared
// by all 8 waves. 512 v_wmma_f32_16x16x4_f32 per wave.
// Grid: B * (C/16) * (NQ/16/8) = 16 blocks x 256 threads.
// ---------------------------------------------------------------------------
__global__ void attn_v_kernel(const float* __restrict__ attn,  // (B*NQ, NKV)
                              const float* __restrict__ kvp,   // (B*NKV, 2C), v=[C,2C)
                              float* __restrict__ qout)        // (B*NQ, C)
{
    __shared__ float sV[256 * 16];   // 256 k x 16 cols (16 KB)

    const int lane = threadIdx.x & 31;
    const int warp = threadIdx.x >> 5;
    int bid = blockIdx.x;
    const int ng = bid % (NQ_ / 16 / 8);  bid /= (NQ_ / 16 / 8);  // n-tile group
    const int ct = bid % (C_ / 16);       bid /= (C_ / 16);
    const int b  = bid;
    const int nt = ng * 8 + warp;
    const int n0 = nt * 16, c0 = ct * 16;
    const int half = lane >> 4, rr = lane & 15;

    const float* Arow = attn + (size_t)(b * NQ_ + n0 + rr) * NKV_;

    v8f acc = {};
    for (int k0 = 0; k0 < NKV_; k0 += 256) {
        __syncthreads();  // previous chunk fully consumed
        {
            const int m = threadIdx.x;  // one key row per thread
            const float* src = kvp + (size_t)(b * NKV_ + k0 + m) * (2 * C_) + C_ + c0;
#pragma unroll
            for (int c4 = 0; c4 < 4; ++c4)
                async_copy16(src + c4 * 4, &sV[m * 16 + c4 * 4]);
        }
        async_wait0();
        __syncthreads();

#pragma unroll 8
        for (int kk = 0; kk < 256; kk += 4) {
            __builtin_prefetch(Arow + k0 + kk + 128, 0, 1);  // global_prefetch_b8
            v2f a, bb;
            a.x = Arow[k0 + kk + 2 * half];
            a.y = Arow[k0 + kk + 2 * half + 1];
            bb.x = sV[(kk + 2 * half) * 16 + rr];
            bb.y = sV[(kk + 2 * half + 1) * 16 + rr];
            acc = __builtin_amdgcn_wmma_f32_16x16x4_f32(
                false, a, false, bb, (short)0, acc, false, false);
        }
    }
#pragma unroll
    for (int g = 0; g < 8; ++g)
        qout[(size_t)(b * NQ_ + n0 + g + 8 * half) * C_ + c0 + rr] = acc[g];
}

// ---------------------------------------------------------------------------
extern "C" void kernel_launch(void* const* d_in, const int* in_sizes, int n_in,
                              void* d_out, int out_size, void* d_ws, size_t ws_size,
                              hipStream_t stream)
{
    const float* q        = (const float*)d_in[0];  // (B,NQ,C)
    const float* q_coord  = (const float*)d_in[1];  // (B,NQ,3)
    const float* kv       = (const float*)d_in[2];  // (B,NKV,ICO)
    const float* kv_coord = (const float*)d_in[3];  // (B,NKV,3)
    const float* Wq       = (const float*)d_in[4];  // (ITERS,C,C)
    const float* Wkv      = (const float*)d_in[5];  // (ITERS,2C,ICO)
    const float* Wdelta   = (const float*)d_in[6];  // (ITERS,C,3)
    const float* Wp       = (const float*)d_in[7];  // (C,C)
    const float* bp       = (const float*)d_in[8];  // (C,)
    float* out = (float*)d_out;

    float* ws  = (float*)d_ws;
    float* qp  = ws;                       // B*NQ*C      = 32768 f (pre-scaled by L2E)
    float* qnx = qp  + B_ * NQ_ * C_;      // B*NQ*C      = 32768 f
    float* kvp = qnx + B_ * NQ_ * C_;      // B*NKV*2C    = 262144 f (k half pre-scaled)
    float* S   = kvp + B_ * NKV_ * 2 * C_; // B*NQ*NKV    = 2097152 f (~8 MB, L2-resident)

    const float* qin = q;
    for (int i = 0; i < ITERS_; ++i) {
        // q projection: 1024x32 @ 32x32, output scaled by log2(e)
        proj_gemm_kernel<<<16, 256, 0, stream>>>(qin, Wq + (size_t)i * C_ * C_,
                                                 nullptr, qp, B_ * NQ_, C_, C_, C_);
        // kv projection: 4096x64 @ 64x64; only k-half (cols < C) scaled
        proj_gemm_kernel<<<128, 256, 0, stream>>>(kv, Wkv + (size_t)i * 2 * C_ * ICO_,
                                                  nullptr, kvp, B_ * NKV_, ICO_, 2 * C_, C_);
        // affinity S: 2 * 32 * 8 = 512 blocks of 256 threads
        affinity_kernel<<<B_ * (NQ_ / 16) * (NKV_ / 256), 256, 0, stream>>>(
            qp, q_coord, kvp, kv_coord, Wdelta + (size_t)i * C_ * 3, S);
        // softmax: 1024 rows, 8 waves per block
        softmax_kernel<<<(B_ * NQ_) / 8, 256, 0, stream>>>(S);
        // attn @ v: 16 blocks, LDS-staged v shared by 8 waves
        attn_v_kernel<<<B_ * (C_ / 16) * (NQ_ / 16 / 8), 256, 0, stream>>>(S, kvp, qnx);
        qin = qnx;
    }
    // final projection with bias, no scaling
    proj_gemm_kernel<<<16, 256, 0, stream>>>(qin, Wp, bp, out, B_ * NQ_, C_, C_, 0);
}